// AngleLoss_11321533792542
// MI455X (gfx1250) — compile-verified
//
#include <hip/hip_runtime.h>

// ---------------------------------------------------------------------------
// AngleLoss (HANDS17 bone-angle loss), MI455X / gfx1250, wave32.
//
// HBM-bound: 264 MB streamed once -> ~11us floor at 23.3 TB/s.
// - Per-wave tiles staged into private LDS with b128 transactions via the
//   CDNA5 async global->LDS pipe (GLOBAL_LOAD_ASYNC_TO_LDS_B128, ASYNCcnt).
// - Stride-63 per-lane rows in LDS: coprime with 64 banks -> conflict-free.
// - Exact-f32 cross-lane reduction via V_WMMA_F32_16X16X4_F32 (ones-B trick).
// - Deterministic two-pass final sum (no float atomics -> replay-stable).
// ---------------------------------------------------------------------------

#define WAVES_PER_BLOCK 4
#define TPB             (WAVES_PER_BLOCK * 32)
#define NBLOCKS         2048
#define ELEMS_PER_WAVE  32
#define FLOATS_PER_ELEM 63                                   // 21 joints * 3
#define TILE_FLOATS     (ELEMS_PER_WAVE * FLOATS_PER_ELEM)   // 2016 used
#define TILE_PAD        2048                                 // float4-friendly
#define VEC4_PER_LANE   16                                   // 16*32*4 = 2048 floats
#define NPAIRS          20

typedef __attribute__((ext_vector_type(2))) float v2f;
typedef __attribute__((ext_vector_type(8))) float v8f;
typedef __attribute__((ext_vector_type(4))) int   v4i;

#if __has_builtin(__builtin_amdgcn_global_load_async_to_lds_b128) && \
    __has_builtin(__builtin_amdgcn_s_wait_asynccnt)
#define HAVE_ASYNC_LDS 1
typedef __attribute__((address_space(1))) v4i as1_v4i;   // global-memory v4i
typedef __attribute__((address_space(3))) v4i as3_v4i;   // LDS v4i
#endif

// Exact f32 wave-sum: A[m][k] holds s in one K slot, 0 in the other; B = ones,
// C = 0 => D[m][n] = s_m + s_{m+16} exactly. Rows 0..7 land in d[0..7] of
// lanes 0-15, rows 8..15 in lanes 16-31; xor-16 shuffle finishes the total.
__device__ __forceinline__ float wave_reduce_wmma(float s) {
    v2f a; a.x = s;    a.y = 0.0f;
    v2f b; b.x = 1.0f; b.y = 1.0f;
    v8f c = {};
    v8f d = __builtin_amdgcn_wmma_f32_16x16x4_f32(
        false, a, false, b, (short)0, c, false, false);
    float t = ((d[0] + d[1]) + (d[2] + d[3])) + ((d[4] + d[5]) + (d[6] + d[7]));
    t += __shfl_xor(t, 16, 32);
    return t;                       // uniform across all 32 lanes
}

__global__ void __launch_bounds__(TPB)
angle_loss_partial(const float* __restrict__ pred,
                   const float* __restrict__ gt,
                   float* __restrict__ part, int B) {
    __shared__ __align__(16) float ldsP[WAVES_PER_BLOCK * TILE_PAD];
    __shared__ __align__(16) float ldsG[WAVES_PER_BLOCK * TILE_PAD];
    __shared__ float wsum[WAVES_PER_BLOCK];

    const int lane  = threadIdx.x & 31;
    const int wave  = threadIdx.x >> 5;
    const int gwave = blockIdx.x * WAVES_PER_BLOCK + wave;
    const int nwave = gridDim.x * WAVES_PER_BLOCK;
    const long long totalF = (long long)B * FLOATS_PER_ELEM;

    float* tP = &ldsP[wave * TILE_PAD];
    float* tG = &ldsG[wave * TILE_PAD];

    constexpr int I0[NPAIRS] = {0,0,0,0,0,1,6,7,2,9,10,3,12,13,4,15,16,5,18,19};
    constexpr int I1[NPAIRS] = {1,2,3,4,5,6,7,8,9,10,11,12,13,14,15,16,17,18,19,20};

    float s = 0.0f;
    const long long step = (long long)nwave * ELEMS_PER_WAVE;

    for (long long base = (long long)gwave * ELEMS_PER_WAVE; base < B; base += step) {
        const long long fbase = base * FLOATS_PER_ELEM;

        // Prefetch next tile of both streams toward GL2 (global_prefetch_b8).
        const long long nfb = fbase + step * FLOATS_PER_ELEM;
        if (nfb < totalF) {
            __builtin_prefetch(&pred[nfb + (long long)lane * 64], 0, 0);
            __builtin_prefetch(&gt  [nfb + (long long)lane * 64], 0, 0);
        }

        const bool full = (fbase + TILE_PAD) <= totalF;   // wide path reads 2048 floats
        if (full) {
            const float4* gp  = reinterpret_cast<const float4*>(pred + fbase);
            const float4* gg  = reinterpret_cast<const float4*>(gt   + fbase);
            float4*       l4P = reinterpret_cast<float4*>(tP);
            float4*       l4G = reinterpret_cast<float4*>(tG);
#if defined(HAVE_ASYNC_LDS)
#pragma unroll
            for (int i = 0; i < VEC4_PER_LANE; ++i) {
                const int k = i * 32 + lane;
                __builtin_amdgcn_global_load_async_to_lds_b128(
                    (as1_v4i*)(gp + k), (as3_v4i*)(l4P + k), 0, 0);
                __builtin_amdgcn_global_load_async_to_lds_b128(
                    (as1_v4i*)(gg + k), (as3_v4i*)(l4G + k), 0, 0);
            }
            __builtin_amdgcn_s_wait_asynccnt(0);   // tile resident in LDS
#else
#pragma unroll
            for (int i = 0; i < VEC4_PER_LANE; ++i) {
                const int k = i * 32 + lane;       // coalesced b128 both sides
                l4P[k] = gp[k];
                l4G[k] = gg[k];
            }
#endif
        } else {
            // Ragged last tile: guarded scalar staging.
            for (int k = 0; k < FLOATS_PER_ELEM; ++k) {
                const long long gi = fbase + (long long)(k * 32 + lane);
                const bool ok = (gi < totalF);
                tP[k * 32 + lane] = ok ? pred[gi] : 0.0f;
                tG[k * 32 + lane] = ok ? gt[gi]   : 0.0f;
            }
        }
        // Wave-private region; same-wave DS ordering (async path: explicit
        // s_wait_asynccnt above) makes the tile safe to read, no barrier.

        const long long elem = base + lane;
        if (elem < B) {
            const float* pu = &tP[lane * FLOATS_PER_ELEM];  // bank-conflict-free
            const float* gu = &tG[lane * FLOATS_PER_ELEM];
            float acc = 0.0f;
#pragma unroll
            for (int p = 0; p < NPAIRS; ++p) {
                const int a3 = I0[p] * 3, b3 = I1[p] * 3;
                const float dx1 = pu[a3]     - pu[b3];
                const float dy1 = pu[a3 + 1] - pu[b3 + 1];
                const float dx2 = gu[a3]     - gu[b3];
                const float dy2 = gu[a3 + 1] - gu[b3 + 1];
                const float dot = dx1 * dx2 + dy1 * dy2;
                const float n1  = dx1 * dx1 + dy1 * dy1;
                const float n2  = dx2 * dx2 + dy2 * dy2;
                acc += 1.0f - fabsf(dot) * rsqrtf(n1 * n2);
            }
            s += acc;   // data dependence forces s_wait_dscnt before back-edge
        }
    }

    // EXEC all-1s here (full blocks, no early returns).
    const float wtot = wave_reduce_wmma(s);
    if (lane == 0) wsum[wave] = wtot;
    __syncthreads();
    if (threadIdx.x == 0) {
        float bsum = 0.0f;
        for (int w = 0; w < WAVES_PER_BLOCK; ++w) bsum += wsum[w];
        part[blockIdx.x] = bsum;
    }
}

__global__ void __launch_bounds__(32)
angle_loss_final(const float* __restrict__ part, int n,
                 float* __restrict__ out, float scale) {
    const int lane = threadIdx.x;
    float s = 0.0f;
    for (int i = lane; i < n; i += 32) s += part[i];   // fixed order: deterministic
    const float t = wave_reduce_wmma(s);
    if (lane == 0) out[0] = t * scale;
}

extern "C" void kernel_launch(void* const* d_in, const int* in_sizes, int n_in,
                              void* d_out, int out_size, void* d_ws, size_t ws_size,
                              hipStream_t stream) {
    const float* pred = (const float*)d_in[0];
    const float* gt   = (const float*)d_in[1];
    const int B = in_sizes[0] / FLOATS_PER_ELEM;     // 524288

    float* part = (float*)d_ws;                      // NBLOCKS floats of scratch

    angle_loss_partial<<<NBLOCKS, TPB, 0, stream>>>(pred, gt, part, B);

    const float scale = (float)(1.0 / ((double)B * (double)NPAIRS));
    angle_loss_final<<<1, 32, 0, stream>>>(part, NBLOCKS, (float*)d_out, scale);
}